// BillboardAllocatorGNN_35871566856905
// MI455X (gfx1250) — compile-verified
//
#include <hip/hip_runtime.h>
#include <hip/hip_bf16.h>
#include <math.h>

#define N_NODES 20000
#define N_EDGES 320000
#define BATCH   4
#define NODE_F  8
#define DIM     128
#define ROWS    (BATCH * N_NODES)          // 80000
#define HN      ((size_t)ROWS * DIM)       // 10,240,000 floats per h buffer
#define NEG_INF_V (-1000000000.0f)
#define NS_BLOCKS  64
#define CTX_CHUNKS 80                      // 250 nodes per chunk

typedef __attribute__((ext_vector_type(2))) float v2f;
typedef __attribute__((ext_vector_type(8))) float v8f;

// ---------------------------------------------------------------------------
// gfx1250 async global->LDS copy helpers (ASYNCcnt-tracked, ISA 15.18.3)
// ---------------------------------------------------------------------------
__device__ __forceinline__ void async_copy16(const void* gptr, void* lptr)
{
    unsigned lds = (unsigned)(uintptr_t)lptr;   // low 32 bits of flat LDS addr
    asm volatile("global_load_async_to_lds_b128 %0, %1, off"
                 :: "v"(lds), "v"(gptr) : "memory");
}
__device__ __forceinline__ void async_wait0()
{
    asm volatile("s_wait_asynccnt 0x0" ::: "memory");
}

// ---------------------------------------------------------------------------
// 1a. Per-feature partial sums / sumsq over node rows (64 blocks).
// ---------------------------------------------------------------------------
__global__ __launch_bounds__(256) void node_stats_partial_kernel(
    const float* __restrict__ nodes, float* __restrict__ part)
{
    __shared__ float rs[NODE_F][256];
    __shared__ float rq[NODE_F][256];
    float s[NODE_F], q[NODE_F];
#pragma unroll
    for (int f = 0; f < NODE_F; ++f) { s[f] = 0.f; q[f] = 0.f; }
    for (int r = blockIdx.x * 256 + threadIdx.x; r < ROWS; r += 256 * NS_BLOCKS) {
        const float* p = nodes + (size_t)r * NODE_F;
#pragma unroll
        for (int f = 0; f < NODE_F; ++f) {
            float v = p[f];
            s[f] += v;
            q[f] += v * v;
        }
    }
#pragma unroll
    for (int f = 0; f < NODE_F; ++f) { rs[f][threadIdx.x] = s[f]; rq[f][threadIdx.x] = q[f]; }
    __syncthreads();
    for (int o = 128; o > 0; o >>= 1) {
        if (threadIdx.x < o) {
#pragma unroll
            for (int f = 0; f < NODE_F; ++f) {
                rs[f][threadIdx.x] += rs[f][threadIdx.x + o];
                rq[f][threadIdx.x] += rq[f][threadIdx.x + o];
            }
        }
        __syncthreads();
    }
    if (threadIdx.x < NODE_F) {
        part[blockIdx.x * 16 + threadIdx.x]          = rs[threadIdx.x][0];
        part[blockIdx.x * 16 + NODE_F + threadIdx.x] = rq[threadIdx.x][0];
    }
}

// 1b. finalize mean / rstd
__global__ __launch_bounds__(32) void node_stats_final_kernel(
    const float* __restrict__ part, float* __restrict__ stats)
{
    __shared__ float tot[16];
    int f = threadIdx.x;
    if (f < 16) {
        float s = 0.f;
        for (int b = 0; b < NS_BLOCKS; ++b) s += part[b * 16 + f];
        tot[f] = s;
    }
    __syncthreads();
    if (f < NODE_F) {
        float m   = tot[f] / (float)ROWS;
        float var = fmaxf(tot[NODE_F + f] / (float)ROWS - m * m, 0.f);
        stats[f]          = m;
        stats[NODE_F + f] = 1.f / (sqrtf(var) + 1e-8f);
    }
}

// ---------------------------------------------------------------------------
// 2. h0 = relu(norm(nodes) @ W_in + b_in)   (K=8, scalar path is fine)
// ---------------------------------------------------------------------------
__global__ __launch_bounds__(256) void embed_kernel(
    const float* __restrict__ nodes, const float* __restrict__ stats,
    const float* __restrict__ Win, const float* __restrict__ bin,
    float* __restrict__ h)
{
    int idx = blockIdx.x * 256 + threadIdx.x;   // ROWS*DIM exactly
    int row = idx >> 7;
    int d   = idx & (DIM - 1);
    const float* p = nodes + (size_t)row * NODE_F;
    float acc = bin[d];
#pragma unroll
    for (int f = 0; f < NODE_F; ++f)
        acc = fmaf((p[f] - stats[f]) * stats[NODE_F + f], Win[f * DIM + d], acc);
    h[idx] = fmaxf(acc, 0.f);
}

// ---------------------------------------------------------------------------
// 3a. zero agg buffer
// ---------------------------------------------------------------------------
__global__ __launch_bounds__(256) void zero_kernel(float4* __restrict__ p)
{
    size_t idx = (size_t)blockIdx.x * 256 + threadIdx.x;  // HN/4 threads
    p[idx] = make_float4(0.f, 0.f, 0.f, 0.f);
}

// ---------------------------------------------------------------------------
// 3b. agg[dst] += h[src]  (segment_sum; block = 8 edges x 32 float4-chunks)
// ---------------------------------------------------------------------------
__global__ __launch_bounds__(256) void scatter_kernel(
    const int* __restrict__ links, const float* __restrict__ h,
    float* __restrict__ agg)
{
    const int blocksPerBatch = N_EDGES / 8;            // 40000
    int b = blockIdx.x / blocksPerBatch;
    int e = (blockIdx.x % blocksPerBatch) * 8 + (threadIdx.x >> 5);
    int c = (threadIdx.x & 31) * 4;
    const int* Lb = links + (size_t)b * 2 * N_EDGES;
    int src = Lb[e];
    int dst = Lb[N_EDGES + e];
    const float4 v = *(const float4*)(h + ((size_t)b * N_NODES + src) * DIM + c);
    float* ap = agg + ((size_t)b * N_NODES + dst) * DIM + c;
    atomicAdd(ap + 0, v.x);
    atomicAdd(ap + 1, v.y);
    atomicAdd(ap + 2, v.z);
    atomicAdd(ap + 3, v.w);
}

// ---------------------------------------------------------------------------
// 3c. h_out = relu(agg @ W_m + h_in @ W_s + b)  -- dual fp32 WMMA GEMM.
//     A tiles (16x128 from agg and h) async-staged into LDS once per block;
//     8 waves each own one 16x16 output tile over the 128 columns.
//     V_WMMA_F32_16X16X4_F32 lane layout:
//       A (16x4): m = lane&15, kOff = 2*(lane>>4), VGPR v -> A[m][k0+kOff+v]
//       B (4x16): n = lane&15, VGPR v -> B[k0+kOff+v][n]
//       C (16x16): VGPR r -> C[r + 8*(lane>>4)][lane&15]
// ---------------------------------------------------------------------------
__global__ __launch_bounds__(256) void mp_gemm_kernel(
    const float* __restrict__ agg, const float* __restrict__ hin,
    const float* __restrict__ Wm, const float* __restrict__ Ws,
    const float* __restrict__ bias, float* __restrict__ hout)
{
    __shared__ float Atile[2][16][DIM];                // 16 KB
    int tid     = threadIdx.x;
    int rowBase = blockIdx.x * 16;

    // stage A: 2 x 16 x 128 floats = 1024 float4; 4 per thread, async to LDS
    for (int i = tid; i < 1024; i += 256) {
        int srcSel = i >> 9;                           // 0: agg, 1: h
        int j      = i & 511;
        int r      = j >> 5;
        int c4     = j & 31;
        const float* src = (srcSel ? hin : agg) + (size_t)(rowBase + r) * DIM + c4 * 4;
        async_copy16(src, &Atile[srcSel][r][c4 * 4]);
    }
    async_wait0();
    __syncthreads();

    int wave    = tid >> 5;
    int lane    = tid & 31;
    int colBase = wave * 16;
    int m    = lane & 15;
    int n    = lane & 15;
    int kOff = (lane >> 4) << 1;

    v8f acc = {};
    for (int k0 = 0; k0 < DIM; k0 += 4) {
        v2f aA = *(const v2f*)&Atile[0][m][k0 + kOff];
        v2f aH = *(const v2f*)&Atile[1][m][k0 + kOff];
        v2f bM, bS;
        bM.x = Wm[(k0 + kOff)     * DIM + colBase + n];
        bM.y = Wm[(k0 + kOff + 1) * DIM + colBase + n];
        bS.x = Ws[(k0 + kOff)     * DIM + colBase + n];
        bS.y = Ws[(k0 + kOff + 1) * DIM + colBase + n];
        acc = __builtin_amdgcn_wmma_f32_16x16x4_f32(false, aA, false, bM,
                                                    (short)0, acc, false, false);
        acc = __builtin_amdgcn_wmma_f32_16x16x4_f32(false, aH, false, bS,
                                                    (short)0, acc, false, false);
    }

    float bv    = bias[colBase + n];
    int rshift  = (lane >> 4) * 8;
    float* outp = hout + (size_t)rowBase * DIM + colBase + n;
#pragma unroll
    for (int r = 0; r < 8; ++r) {
        float v = acc[r] + bv;
        outp[(size_t)(r + rshift) * DIM] = fmaxf(v, 0.f);
    }
}

// ---------------------------------------------------------------------------
// 4. ad normalization + a = relu(ad_n @ W_ad + b_ad)   (1 block, 512 threads)
// ---------------------------------------------------------------------------
__global__ __launch_bounds__(512) void ad_kernel(
    const float* __restrict__ ad, const float* __restrict__ Wad,
    const float* __restrict__ bad, float* __restrict__ a)
{
    __shared__ float mean[NODE_F], rstd[NODE_F];
    if (threadIdx.x < NODE_F) {
        float s = 0.f, q = 0.f;
        for (int b = 0; b < BATCH; ++b) {
            float v = ad[b * NODE_F + threadIdx.x];
            s += v; q += v * v;
        }
        float mn  = s / (float)BATCH;
        float var = fmaxf(q / (float)BATCH - mn * mn, 0.f);
        mean[threadIdx.x] = mn;
        rstd[threadIdx.x] = 1.f / (sqrtf(var) + 1e-8f);
    }
    __syncthreads();
    int b = threadIdx.x >> 7;
    int d = threadIdx.x & (DIM - 1);
    float acc = bad[d];
#pragma unroll
    for (int f = 0; f < NODE_F; ++f)
        acc = fmaf((ad[b * NODE_F + f] - mean[f]) * rstd[f], Wad[f * DIM + d], acc);
    a[b * DIM + d] = fmaxf(acc, 0.f);
}

// ---------------------------------------------------------------------------
// 5. Policy head: logits = relu([h | a] @ W_p + b_p) . w_out, masked.
//    h rows async-staged to LDS; a[b] staged to LDS; WMMA over K=256;
//    fused w_out dot via shfl_xor + LDS cross-wave reduction.
// ---------------------------------------------------------------------------
__global__ __launch_bounds__(256) void policy_kernel(
    const float* __restrict__ h, const float* __restrict__ a,
    const float* __restrict__ Wp, const float* __restrict__ bp,
    const float* __restrict__ wout, const unsigned char* __restrict__ mask,
    float* __restrict__ logits)
{
    __shared__ float Htile[16][DIM];                   // 8 KB
    __shared__ float aSh[DIM];
    __shared__ float lsum[16];
    int tid     = threadIdx.x;
    int rowBase = blockIdx.x * 16;          // never crosses a batch (20000 % 16 == 0)
    int b       = rowBase / N_NODES;

    for (int i = tid; i < 512; i += 256) {             // 16x128 floats = 512 float4
        int r  = i >> 5;
        int c4 = i & 31;
        async_copy16(h + (size_t)(rowBase + r) * DIM + c4 * 4, &Htile[r][c4 * 4]);
    }
    if (tid < DIM) aSh[tid] = a[(size_t)b * DIM + tid];
    if (tid < 16)  lsum[tid] = 0.f;
    async_wait0();
    __syncthreads();

    int wave    = tid >> 5;
    int lane    = tid & 31;
    int colBase = wave * 16;
    int m    = lane & 15;
    int n    = lane & 15;
    int kOff = (lane >> 4) << 1;

    v8f acc = {};
    for (int k0 = 0; k0 < 2 * DIM; k0 += 4) {
        v2f av = (k0 < DIM) ? *(const v2f*)&Htile[m][k0 + kOff]
                            : *(const v2f*)&aSh[(k0 - DIM) + kOff];
        v2f bv;
        bv.x = Wp[(k0 + kOff)     * DIM + colBase + n];
        bv.y = Wp[(k0 + kOff + 1) * DIM + colBase + n];
        acc = __builtin_amdgcn_wmma_f32_16x16x4_f32(false, av, false, bv,
                                                    (short)0, acc, false, false);
    }

    float wo = wout[colBase + n];
    float bb = bp[colBase + n];
    int rshift = (lane >> 4) * 8;
#pragma unroll
    for (int r = 0; r < 8; ++r) {
        float v = fmaxf(acc[r] + bb, 0.f) * wo;
#pragma unroll
        for (int o = 1; o < 16; o <<= 1)
            v += __shfl_xor(v, o, 32);
        if (n == 0) atomicAdd(&lsum[r + rshift], v);
    }
    __syncthreads();
    if (tid < 16) {
        int row = rowBase + tid;
        logits[row] = mask[row] ? lsum[tid] : NEG_INF_V;
    }
}

// ---------------------------------------------------------------------------
// 6. Per-batch softmax stats (max, log-sum-exp)
// ---------------------------------------------------------------------------
__global__ __launch_bounds__(256) void softmax_stats_kernel(
    const float* __restrict__ logits, float* __restrict__ red)
{
    __shared__ float s[256];
    int b = blockIdx.x;
    const float* lg = logits + (size_t)b * N_NODES;
    float mx = -3.4e38f;
    for (int i = threadIdx.x; i < N_NODES; i += 256) mx = fmaxf(mx, lg[i]);
    s[threadIdx.x] = mx;
    __syncthreads();
    for (int o = 128; o > 0; o >>= 1) {
        if (threadIdx.x < o) s[threadIdx.x] = fmaxf(s[threadIdx.x], s[threadIdx.x + o]);
        __syncthreads();
    }
    mx = s[0];
    __syncthreads();
    float se = 0.f;
    for (int i = threadIdx.x; i < N_NODES; i += 256) se += expf(lg[i] - mx);
    s[threadIdx.x] = se;
    __syncthreads();
    for (int o = 128; o > 0; o >>= 1) {
        if (threadIdx.x < o) s[threadIdx.x] += s[threadIdx.x + o];
        __syncthreads();
    }
    if (threadIdx.x == 0) {
        red[b * 2]     = mx;
        red[b * 2 + 1] = logf(s[0]);
    }
}

// ---------------------------------------------------------------------------
// 7. log_probs -> d_out, attn -> ws
// ---------------------------------------------------------------------------
__global__ __launch_bounds__(256) void logprob_kernel(
    const float* __restrict__ logits, const float* __restrict__ red,
    float* __restrict__ out, float* __restrict__ attn)
{
    int idx = blockIdx.x * 256 + threadIdx.x;
    if (idx >= ROWS) return;
    int b = idx / N_NODES;
    float lp = logits[idx] - red[b * 2] - red[b * 2 + 1];
    out[idx]  = lp;
    attn[idx] = expf(lp);
}

// ---------------------------------------------------------------------------
// 8a. context partial: chunked over nodes for parallelism (deterministic)
// ---------------------------------------------------------------------------
__global__ __launch_bounds__(128) void context_partial_kernel(
    const float* __restrict__ attn, const float* __restrict__ h,
    float* __restrict__ part)
{
    int blk = blockIdx.x;                 // BATCH * CTX_CHUNKS
    int b   = blk / CTX_CHUNKS;
    int ch  = blk % CTX_CHUNKS;
    int d   = threadIdx.x;
    const int span = N_NODES / CTX_CHUNKS;      // 250
    int n0  = ch * span;
    const float* hb = h + ((size_t)b * N_NODES + n0) * DIM;
    const float* ab = attn + (size_t)b * N_NODES + n0;
    float s = 0.f;
    for (int nn = 0; nn < span; ++nn)
        s = fmaf(ab[nn], hb[(size_t)nn * DIM + d], s);
    part[(size_t)blk * DIM + d] = s;
}

// 8b. context reduce
__global__ __launch_bounds__(128) void context_reduce_kernel(
    const float* __restrict__ part, float* __restrict__ ctx)
{
    int b = blockIdx.x;
    int d = threadIdx.x;
    float s = 0.f;
    for (int c = 0; c < CTX_CHUNKS; ++c)
        s += part[((size_t)b * CTX_CHUNKS + c) * DIM + d];
    ctx[b * DIM + d] = s;
}

// ---------------------------------------------------------------------------
// 9. value[b] = relu([ctx | a] @ W_v + b_v) . w_vo
// ---------------------------------------------------------------------------
__global__ __launch_bounds__(128) void value_kernel(
    const float* __restrict__ ctx, const float* __restrict__ a,
    const float* __restrict__ Wv, const float* __restrict__ bv,
    const float* __restrict__ wvo, float* __restrict__ out)
{
    __shared__ float s[128];
    int b = blockIdx.x;
    int d = threadIdx.x;
    float acc = bv[d];
    for (int k = 0; k < DIM; ++k)
        acc = fmaf(ctx[b * DIM + k], Wv[k * DIM + d], acc);
    for (int k = 0; k < DIM; ++k)
        acc = fmaf(a[b * DIM + k], Wv[(DIM + k) * DIM + d], acc);
    s[d] = fmaxf(acc, 0.f) * wvo[d];
    __syncthreads();
    for (int o = 64; o > 0; o >>= 1) {
        if (d < o) s[d] += s[d + o];
        __syncthreads();
    }
    if (d == 0) out[ROWS + b] = s[0];
}

// ---------------------------------------------------------------------------
extern "C" void kernel_launch(void* const* d_in, const int* in_sizes, int n_in,
                              void* d_out, int out_size, void* d_ws, size_t ws_size,
                              hipStream_t stream)
{
    const float*         nodes = (const float*)d_in[0];
    const int*           links = (const int*)d_in[1];
    const unsigned char* mask  = (const unsigned char*)d_in[2];
    const float*         adp   = (const float*)d_in[3];
    const float* Win = (const float*)d_in[4];
    const float* bin = (const float*)d_in[5];
    const float* Wm1 = (const float*)d_in[6];
    const float* Ws1 = (const float*)d_in[7];
    const float* b1  = (const float*)d_in[8];
    const float* Wm2 = (const float*)d_in[9];
    const float* Ws2 = (const float*)d_in[10];
    const float* b2  = (const float*)d_in[11];
    const float* Wad = (const float*)d_in[12];
    const float* bad = (const float*)d_in[13];
    const float* Wp  = (const float*)d_in[14];
    const float* bp  = (const float*)d_in[15];
    const float* wot = (const float*)d_in[16];
    const float* Wv  = (const float*)d_in[17];
    const float* bvv = (const float*)d_in[18];
    const float* wvo = (const float*)d_in[19];

    float* ws      = (float*)d_ws;
    float* h0      = ws;                   // HN
    float* h1      = ws + HN;              // HN
    float* agg     = ws + 2 * HN;          // HN
    float* stats   = ws + 3 * HN;          // 16
    float* a       = stats + 16;           // BATCH*DIM
    float* logits  = a + BATCH * DIM;      // ROWS
    float* attn    = logits + ROWS;        // ROWS
    float* red     = attn + ROWS;          // 2*BATCH
    float* ctx     = red + 2 * BATCH;      // BATCH*DIM
    float* nsPart  = ctx + BATCH * DIM;    // NS_BLOCKS*16
    float* ctxPart = nsPart + NS_BLOCKS * 16; // BATCH*CTX_CHUNKS*DIM
    float* out     = (float*)d_out;

    // 1-2. normalize + embed
    node_stats_partial_kernel<<<NS_BLOCKS, 256, 0, stream>>>(nodes, nsPart);
    node_stats_final_kernel<<<1, 32, 0, stream>>>(nsPart, stats);
    embed_kernel<<<(int)(HN / 256), 256, 0, stream>>>(nodes, stats, Win, bin, h0);

    // 3. message passing layer 1: h0 -> h1
    zero_kernel<<<(int)(HN / 4 / 256), 256, 0, stream>>>((float4*)agg);
    scatter_kernel<<<BATCH * (N_EDGES / 8), 256, 0, stream>>>(links, h0, agg);
    mp_gemm_kernel<<<ROWS / 16, 256, 0, stream>>>(agg, h0, Wm1, Ws1, b1, h1);

    // 3'. message passing layer 2: h1 -> h0 (h0 is the final h)
    zero_kernel<<<(int)(HN / 4 / 256), 256, 0, stream>>>((float4*)agg);
    scatter_kernel<<<BATCH * (N_EDGES / 8), 256, 0, stream>>>(links, h1, agg);
    mp_gemm_kernel<<<ROWS / 16, 256, 0, stream>>>(agg, h1, Wm2, Ws2, b2, h0);

    // 4. ad embedding
    ad_kernel<<<1, 512, 0, stream>>>(adp, Wad, bad, a);

    // 5. policy head
    policy_kernel<<<ROWS / 16, 256, 0, stream>>>(h0, a, Wp, bp, wot, mask, logits);

    // 6-7. masked log-softmax
    softmax_stats_kernel<<<BATCH, 256, 0, stream>>>(logits, red);
    logprob_kernel<<<(ROWS + 255) / 256, 256, 0, stream>>>(logits, red, out, attn);

    // 8-9. attention context + value head
    context_partial_kernel<<<BATCH * CTX_CHUNKS, 128, 0, stream>>>(attn, h0, ctxPart);
    context_reduce_kernel<<<BATCH, 128, 0, stream>>>(ctxPart, ctx);
    value_kernel<<<BATCH, 128, 0, stream>>>(ctx, a, Wv, bvv, wvo, out);
}